// CoGNNLayer_47605417509015
// MI455X (gfx1250) — compile-verified
//
#include <hip/hip_runtime.h>
#include <hip/hip_bf16.h>
#include <math.h>

typedef __attribute__((ext_vector_type(2))) float v2f;
typedef __attribute__((ext_vector_type(8))) float v8f;

#define D 128
#define EF 16

static __device__ __forceinline__ void atomic_add_f32(float* p, float v) {
    __hip_atomic_fetch_add(p, v, __ATOMIC_RELAXED, __HIP_MEMORY_SCOPE_AGENT);
}

// ---------------- 1. edge weights + degree accumulation ----------------
__global__ __launch_bounds__(256) void k_edge_weight(const float* __restrict__ ea,
                                                     const int* __restrict__ dst,
                                                     int E, float* __restrict__ ew,
                                                     float* __restrict__ deg) {
    int e = blockIdx.x * 256 + threadIdx.x;
    if (e >= E) return;
    const float4* p = (const float4*)(ea + (size_t)e * EF);
    float s = 0.f;
#pragma unroll
    for (int i = 0; i < 4; ++i) {
        float4 v = p[i];
        s += v.x * v.x + v.y * v.y + v.z * v.z + v.w * v.w;
    }
    float w = sqrtf(s);
    ew[e] = w;
    atomic_add_f32(&deg[dst[e]], w);
}

// ---------------- 2. self-loop edge feature: relu(1_vec @ Wp + bp) ------
__global__ void k_f0(const float* __restrict__ Wp, const float* __restrict__ bp,
                     float* __restrict__ f0) {
    int c = threadIdx.x; // 128 threads
    float s = bp[c];
#pragma unroll
    for (int k = 0; k < EF; ++k) s += Wp[k * D + c];
    f0[c] = fmaxf(s, 0.f);
}

// ---------------- 3. dinv = rsqrt(deg + 1) ------------------------------
__global__ __launch_bounds__(256) void k_dinv(const float* __restrict__ deg,
                                              float* __restrict__ dinv, int N) {
    int i = blockIdx.x * 256 + threadIdx.x;
    if (i >= N) return;
    dinv[i] = rsqrtf(deg[i] + 1.0f); // >= 1 always (self loop)
}

// ---------------- 4. layernorm + h = xn @ W_gcn (WMMA f32 16x16x4) ------
__global__ __launch_bounds__(256) void k_ln_gemm(const float* __restrict__ x,
                                                 const float* __restrict__ W,
                                                 const float* __restrict__ g,
                                                 const float* __restrict__ b,
                                                 float* __restrict__ h, int N) {
    __shared__ float A[16][D + 4];
    __shared__ float red[16][16];
    __shared__ float red2[16][16];
    __shared__ float mu[16], rs[16];

    int tid = threadIdx.x;
    int row = tid >> 4;   // 0..15
    int l16 = tid & 15;
    int row0 = blockIdx.x * 16;
    int grow = row0 + row; if (grow >= N) grow = N - 1;  // clamp, keep EXEC full

    const float* xr = x + (size_t)grow * D;
    float v[8];
    float s = 0.f, s2 = 0.f;
#pragma unroll
    for (int j = 0; j < 8; ++j) {
        float t = xr[l16 + j * 16];
        v[j] = t; s += t; s2 += t * t;
    }
    red[row][l16] = s; red2[row][l16] = s2;
    __syncthreads();
    if (l16 == 0) {
        float ts = 0.f, ts2 = 0.f;
#pragma unroll
        for (int j = 0; j < 16; ++j) { ts += red[row][j]; ts2 += red2[row][j]; }
        float m = ts * (1.f / 128.f);
        float var = ts2 * (1.f / 128.f) - m * m;
        mu[row] = m; rs[row] = rsqrtf(var + 1e-5f);
    }
    __syncthreads();
    float m = mu[row], r = rs[row];
#pragma unroll
    for (int j = 0; j < 8; ++j) {
        int c = l16 + j * 16;
        A[row][c] = (v[j] - m) * r * g[c] + b[c];
    }
    __syncthreads();

    // each wave: one 16x16 output tile
    int wave = tid >> 5, lane = tid & 31;
    int n0 = wave * 16;
    int lrow = lane & 15;
    int khalf = (lane >> 4) * 2; // 0 for lanes 0-15, 2 for 16-31
    v8f acc = {};
    for (int k = 0; k < D; k += 4) {
        v2f af, bf;
        af.x = A[lrow][k + khalf];
        af.y = A[lrow][k + khalf + 1];
        bf.x = W[(size_t)(k + khalf) * D + n0 + lrow];
        bf.y = W[(size_t)(k + khalf + 1) * D + n0 + lrow];
        acc = __builtin_amdgcn_wmma_f32_16x16x4_f32(false, af, false, bf,
                                                    (short)0, acc, false, false);
    }
    int rbase = (lane < 16) ? 0 : 8;
#pragma unroll
    for (int i = 0; i < 8; ++i) {
        int rr = row0 + rbase + i;
        if (rr < N) h[(size_t)rr * D + n0 + lrow] = acc[i];
    }
}

// ---------------- 5. node init: out = dinv^2 * h ; aggr = f0 ------------
__global__ __launch_bounds__(256) void k_init_nodes(const float* __restrict__ h,
                                                    const float* __restrict__ dinv,
                                                    const float* __restrict__ f0,
                                                    float* __restrict__ outg,
                                                    float* __restrict__ aggr, int N) {
    long idx = (long)blockIdx.x * 256 + threadIdx.x;
    if (idx >= (long)N * D) return;
    int i = (int)(idx >> 7);
    int c = (int)(idx & 127);
    float di = dinv[i];
    outg[idx] = di * di * h[idx];
    aggr[idx] = f0[c];
}

// ---------------- 6. fused edge projection + GCN scatter ----------------
__global__ __launch_bounds__(256) void k_edge_scatter(const float* __restrict__ ea,
                                                      const int* __restrict__ src,
                                                      const int* __restrict__ dst,
                                                      const float* __restrict__ ew,
                                                      const float* __restrict__ dinv,
                                                      const float* __restrict__ h,
                                                      const float* __restrict__ Wp,
                                                      const float* __restrict__ bp,
                                                      float* __restrict__ outg,
                                                      float* __restrict__ aggr, int E) {
    __shared__ float sa[2][EF];
    int tid = threadIdx.x;
    int ge = tid >> 7;          // 0/1: edge within block
    int c = tid & 127;          // channel
    long e = (long)blockIdx.x * 2 + ge;
    bool valid = e < E;
    int s = 0, d = 0; float w = 0.f;
    if (valid) { s = src[e]; d = dst[e]; w = ew[e]; }
    if (valid && c < EF) sa[ge][c] = ea[e * EF + c];
    __syncthreads();
    if (!valid) return;
    // edge feature for channel c: relu(ea . Wp[:,c] + bp[c])
    float f = bp[c];
#pragma unroll
    for (int k = 0; k < EF; ++k) f += sa[ge][k] * Wp[k * D + c];
    f = fmaxf(f, 0.f);
    atomic_add_f32(&aggr[(size_t)d * D + c], f);
    float nrm = dinv[s] * w * dinv[d];
    atomic_add_f32(&outg[(size_t)d * D + c], nrm * h[(size_t)s * D + c]);
}

// ---------------- 7. gate GEMM (WMMA, K=256) + sigmoid combine + relu ----
__global__ __launch_bounds__(256) void k_combine(const float* __restrict__ outg,
                                                 const float* __restrict__ aggr,
                                                 const float* __restrict__ b_gcn,
                                                 const float* __restrict__ Wc,
                                                 const float* __restrict__ bc,
                                                 float* __restrict__ out, int N) {
    __shared__ float A[16][2 * D + 4]; // [16][260]: cols 0..127 = out+b, 128..255 = aggr
    int tid = threadIdx.x;
    int row = tid >> 4, l16 = tid & 15;
    int row0 = blockIdx.x * 16;
    int grow = row0 + row; if (grow >= N) grow = N - 1;
    const float* op = outg + (size_t)grow * D;
    const float* ap = aggr + (size_t)grow * D;
#pragma unroll
    for (int j = 0; j < 8; ++j) {
        int cix = l16 + j * 16;
        A[row][cix] = op[cix] + b_gcn[cix];
        A[row][D + cix] = ap[cix];
    }
    __syncthreads();

    int wave = tid >> 5, lane = tid & 31;
    int n0 = wave * 16;
    int lrow = lane & 15;
    int khalf = (lane >> 4) * 2;
    v8f acc = {};
    for (int k = 0; k < 2 * D; k += 4) {
        v2f af, bf;
        af.x = A[lrow][k + khalf];
        af.y = A[lrow][k + khalf + 1];
        bf.x = Wc[(size_t)(k + khalf) * D + n0 + lrow];
        bf.y = Wc[(size_t)(k + khalf + 1) * D + n0 + lrow];
        acc = __builtin_amdgcn_wmma_f32_16x16x4_f32(false, af, false, bf,
                                                    (short)0, acc, false, false);
    }
    int cc = n0 + lrow;
    int rbase = (lane < 16) ? 0 : 8;
    float bcv = bc[cc];
#pragma unroll
    for (int i = 0; i < 8; ++i) {
        int r = rbase + i;
        float gte = 1.f / (1.f + __expf(-(acc[i] + bcv)));
        float ob = A[r][cc];
        float ag = A[r][D + cc];
        float val = gte * ob + (1.f - gte) * ag;
        int rr = row0 + r;
        if (rr < N) out[(size_t)rr * D + cc] = fmaxf(val, 0.f);
    }
}

extern "C" void kernel_launch(void* const* d_in, const int* in_sizes, int n_in,
                              void* d_out, int out_size, void* d_ws, size_t ws_size,
                              hipStream_t stream) {
    const float* x         = (const float*)d_in[0];
    const float* edge_attr = (const float*)d_in[1];
    const int*   edge_idx  = (const int*)d_in[2];
    const float* W_gcn     = (const float*)d_in[3];
    const float* b_gcn     = (const float*)d_in[4];
    const float* ln_g      = (const float*)d_in[5];
    const float* ln_b      = (const float*)d_in[6];
    const float* Wp        = (const float*)d_in[7];
    const float* bp        = (const float*)d_in[8];
    const float* Wc        = (const float*)d_in[9];
    const float* bc        = (const float*)d_in[10];

    int N = in_sizes[0] / D;
    int E = in_sizes[1] / EF;
    const int* src = edge_idx;
    const int* dst = edge_idx + E;

    float* ws   = (float*)d_ws;
    float* ew   = ws;                     // E
    float* deg  = ew + E;                 // N
    float* dinv = deg + N;                // N
    float* f0   = dinv + N;               // 128
    float* h    = f0 + D;                 // N*D
    float* outg = h + (size_t)N * D;      // N*D
    float* aggr = outg + (size_t)N * D;   // N*D

    hipMemsetAsync(deg, 0, (size_t)N * sizeof(float), stream);
    k_edge_weight<<<(E + 255) / 256, 256, 0, stream>>>(edge_attr, dst, E, ew, deg);
    k_f0<<<1, 128, 0, stream>>>(Wp, bp, f0);
    k_dinv<<<(N + 255) / 256, 256, 0, stream>>>(deg, dinv, N);
    k_ln_gemm<<<(N + 15) / 16, 256, 0, stream>>>(x, W_gcn, ln_g, ln_b, h, N);
    long total = (long)N * D;
    k_init_nodes<<<(int)((total + 255) / 256), 256, 0, stream>>>(h, dinv, f0, outg, aggr, N);
    k_edge_scatter<<<(E + 1) / 2, 256, 0, stream>>>(edge_attr, src, dst, ew, dinv, h,
                                                    Wp, bp, outg, aggr, E);
    k_combine<<<(N + 15) / 16, 256, 0, stream>>>(outg, aggr, b_gcn, Wc, bc,
                                                 (float*)d_out, N);
}